// LIFWeightedMSELoss_24335284699236
// MI455X (gfx1250) — compile-verified
//
#include <hip/hip_runtime.h>

#define N_BINS 256
#define HIST_COPIES 8          // one privatized histogram per wave32 (256 thr = 8 waves)
#define BLOCK_THREADS 256
#define MAX_BLOCKS 2048

static __device__ __forceinline__ int sdf_bin(float t) {
    const float SDF_MIN = -7.0f;
    const float SDF_MAX =  7.0f;
    const float SCALE   = 1.0f / 14.0f;           // 1/(max-min), matches reference
    float c    = fminf(fmaxf(t, SDF_MIN), SDF_MAX);
    float unit = (c - SDF_MIN) * SCALE;
    int idx = __float2int_rn(unit * 255.0f);      // round-half-even, like jnp.round
    idx = idx < 0 ? 0 : (idx > 255 ? 255 : idx);  // fp-safety clamp
    return idx;
}

// ---------------------------------------------------------------------------
// Kernel 0: zero the 2 KB workspace (counts u32[256] + sse f32[256]) each call
// ---------------------------------------------------------------------------
__global__ __launch_bounds__(BLOCK_THREADS) void ws_init_kernel(unsigned* __restrict__ ws) {
    ws[threadIdx.x]          = 0u;   // counts
    ws[threadIdx.x + N_BINS] = 0u;   // sse (0.0f bit pattern)
}

// ---------------------------------------------------------------------------
// Kernel 1: fused single pass — per-wave LDS histograms of {count, sum d^2}
// reads y_pred + y_true exactly once (256 MB total -> HBM-bound ~11us)
// ---------------------------------------------------------------------------
__global__ __launch_bounds__(BLOCK_THREADS) void hist_sse_kernel(
        const float* __restrict__ y_pred,
        const float* __restrict__ y_true,
        unsigned* __restrict__ g_cnt,
        float*    __restrict__ g_sse,
        long long n) {
    __shared__ unsigned s_cnt[HIST_COPIES * N_BINS];
    __shared__ float    s_sse[HIST_COPIES * N_BINS];

    const int tid  = threadIdx.x;
    const int wave = tid >> 5;                     // wave32
    unsigned* my_cnt = &s_cnt[wave * N_BINS];
    float*    my_sse = &s_sse[wave * N_BINS];

    for (int b = tid; b < HIST_COPIES * N_BINS; b += BLOCK_THREADS) {
        s_cnt[b] = 0u;
        s_sse[b] = 0.0f;
    }
    __syncthreads();

    const long long nvec   = n >> 2;               // float4 count
    const long long stride = (long long)gridDim.x * BLOCK_THREADS;
    const float4* yp4 = (const float4*)y_pred;
    const float4* yt4 = (const float4*)y_true;

    for (long long i = (long long)blockIdx.x * BLOCK_THREADS + tid; i < nvec; i += stride) {
        // speculative prefetch one grid-stride ahead (global_prefetch_b8;
        // OOB translations are silently dropped per ISA)
        __builtin_prefetch(&yt4[i + stride], 0, 0);
        __builtin_prefetch(&yp4[i + stride], 0, 0);

        float4 t = yt4[i];
        float4 p = yp4[i];
        float tv[4] = {t.x, t.y, t.z, t.w};
        float pv[4] = {p.x, p.y, p.z, p.w};
        #pragma unroll
        for (int k = 0; k < 4; ++k) {
            int   idx = sdf_bin(tv[k]);
            float d   = pv[k] - tv[k];
            atomicAdd(&my_cnt[idx], 1u);       // ds_add_u32
            atomicAdd(&my_sse[idx], d * d);    // ds_add_f32
        }
    }

    // scalar tail (n not multiple of 4) — handled by block 0
    if (blockIdx.x == 0) {
        long long base = nvec << 2;
        if (base + tid < n) {
            float t = y_true[base + tid];
            float p = y_pred[base + tid];
            int   idx = sdf_bin(t);
            float d   = p - t;
            atomicAdd(&my_cnt[idx], 1u);
            atomicAdd(&my_sse[idx], d * d);
        }
    }
    __syncthreads();

    // merge the 8 wave-private copies; one global atomic pair per touched bin
    unsigned c = 0;
    float    s = 0.0f;
    #pragma unroll
    for (int w = 0; w < HIST_COPIES; ++w) {
        c += s_cnt[w * N_BINS + tid];
        s += s_sse[w * N_BINS + tid];
    }
    if (c) {
        atomicAdd(&g_cnt[tid], c);
        atomicAdd(&g_sse[tid], s);
    }
}

// ---------------------------------------------------------------------------
// Kernel 2: finalize with V_WMMA_F32_16X16X4_F32 (wave32, EXEC all ones).
// total = sum_b sse[b] * (1/log1p(eps + cnt[b]/N)).
// All-ones A (16x4) => D = A*B + C accumulates column sums of B; four
// accumulating WMMAs reduce all 256 per-bin products.
// ---------------------------------------------------------------------------
typedef __attribute__((ext_vector_type(2))) float v2f;
typedef __attribute__((ext_vector_type(8))) float v8f;

__global__ __launch_bounds__(32) void finalize_kernel(
        const unsigned* __restrict__ g_cnt,
        const float*    __restrict__ g_sse,
        float* __restrict__ out,
        long long n) {
    const int   lane  = threadIdx.x;              // 0..31, one full wave32
    const float inv_n = 1.0f / (float)n;
    const float EPS   = 0.02f;

    float prod[8];
    #pragma unroll
    for (int j = 0; j < 8; ++j) {
        int   b    = j * 32 + lane;
        float freq = (float)g_cnt[b] * inv_n;
        float lut  = 1.0f / log1pf(EPS + freq);
        prod[j]    = g_sse[b] * lut;
    }

    v2f a; a.x = 1.0f; a.y = 1.0f;                // all-ones 16x4 A-matrix
    v8f c = {};
    #pragma unroll
    for (int i = 0; i < 4; ++i) {
        v2f b; b.x = prod[2 * i]; b.y = prod[2 * i + 1];   // 64 values per step
        // v_wmma_f32_16x16x4_f32: D = A*B + C (accumulate)
        c = __builtin_amdgcn_wmma_f32_16x16x4_f32(
                /*neg_a=*/false, a, /*neg_b=*/false, b,
                /*c_mod=*/(short)0, c, /*reuse_a=*/false, /*reuse_b=*/false);
    }

    // every row of D is identical = column sums; sum N=0..15 (lanes 0..15 of c[0])
    float colsum = c[0];
    float total  = 0.0f;
    #pragma unroll
    for (int l = 0; l < 16; ++l) total += __shfl(colsum, l, 32);

    if (lane == 0) out[0] = total * inv_n;        // reduction == 'mean'
}

// ---------------------------------------------------------------------------
extern "C" void kernel_launch(void* const* d_in, const int* in_sizes, int n_in,
                              void* d_out, int out_size, void* d_ws, size_t ws_size,
                              hipStream_t stream) {
    const float* y_pred = (const float*)d_in[0];
    const float* y_true = (const float*)d_in[1];
    long long n = (long long)in_sizes[0];

    unsigned* g_cnt = (unsigned*)d_ws;
    float*    g_sse = (float*)((char*)d_ws + N_BINS * sizeof(unsigned));

    ws_init_kernel<<<1, BLOCK_THREADS, 0, stream>>>(g_cnt);

    long long nvec = n >> 2;
    long long want = (nvec + BLOCK_THREADS - 1) / BLOCK_THREADS;
    int blocks = (int)(want < 1 ? 1 : (want > MAX_BLOCKS ? MAX_BLOCKS : want));

    hist_sse_kernel<<<blocks, BLOCK_THREADS, 0, stream>>>(y_pred, y_true, g_cnt, g_sse, n);
    finalize_kernel<<<1, 32, 0, stream>>>(g_cnt, g_sse, (float*)d_out, n);
}